// CRF_18451179504480
// MI455X (gfx1250) — compile-verified
//
#include <hip/hip_runtime.h>
#include <hip/hip_bf16.h>

#define K_TAGS 128
#define B_SZ   256
#define T_LEN  512
#define NEGV   (-10000.0f)
#define ROWPAD 133   // 133 mod 64 = 5 -> conflict-free strided lane access

typedef __attribute__((ext_vector_type(2))) float v2f;
typedef __attribute__((ext_vector_type(8))) float v8f;

__global__ __launch_bounds__(256) void crf_forward_kernel(
    const float* __restrict__ h,      // (B, T, K)
    const float* __restrict__ mask,   // (B, T)
    const float* __restrict__ trans,  // (K, K)
    float* __restrict__ out)          // (B,)
{
  __shared__ float s_score[16][ROWPAD];
  __shared__ float s_pexp [16][ROWPAD];
  __shared__ float s_rowmax[16];

  const int tid   = threadIdx.x;
  const int lane  = tid & 31;
  const int w     = tid >> 5;           // wave 0..7, owns tags [16w, 16w+16)
  const int bbase = blockIdx.x * 16;    // 16 batch rows per block

  const int nloc = lane & 15;
  const int hi   = (lane >> 4) & 1;     // lanes 16..31 hold K+2 / M+8 halves
  const int koff = hi * 2;
  const int jcol = w * 16 + nloc;       // this lane's output tag column
  const int mrow = hi * 8;              // accumulator row base for this lane

  // ---- one-time: B fragments B[k][n] = exp(trans[n_global][k]) ----
  v2f bfrag[32];
#pragma unroll
  for (int ks = 0; ks < 32; ++ks) {
    const int k0 = ks * 4 + koff;
    v2f bf;
    bf.x = __expf(trans[jcol * K_TAGS + k0]);
    bf.y = __expf(trans[jcol * K_TAGS + k0 + 1]);
    bfrag[ks] = bf;
  }

  // ---- init: score = NEG everywhere except SOS (tag 1) = 0 ----
  float oldsc[8];
#pragma unroll
  for (int v = 0; v < 8; ++v) oldsc[v] = (jcol == 1) ? 0.0f : NEGV;
  for (int e = tid; e < 16 * K_TAGS; e += 256) {
    const int r = e >> 7, j = e & 127;
    s_score[r][j] = (j == 1) ? 0.0f : NEGV;
  }
  __syncthreads();

  for (int t = 0; t < T_LEN; ++t) {
    // phase 1: per-batch row max over 128 tags (wave w -> rows 2w, 2w+1)
#pragma unroll
    for (int rr = 0; rr < 2; ++rr) {
      const int r = w * 2 + rr;
      float mx = s_score[r][lane];
      mx = fmaxf(mx, s_score[r][lane + 32]);
      mx = fmaxf(mx, s_score[r][lane + 64]);
      mx = fmaxf(mx, s_score[r][lane + 96]);
      for (int off = 16; off > 0; off >>= 1)
        mx = fmaxf(mx, __shfl_xor(mx, off, 32));
      if (lane == 0) s_rowmax[r] = mx;
    }
    __syncthreads();

    // phase 2: P = exp(score - rowmax)
    for (int e = tid; e < 16 * K_TAGS; e += 256) {
      const int r = e >> 7, j = e & 127;
      s_pexp[r][j] = __expf(s_score[r][j] - s_rowmax[r]);
    }
    __syncthreads();

    // phase 3: New = P (16x128) x E^T (128x16 slice) via f32 WMMA
    v8f acc = {0.f, 0.f, 0.f, 0.f, 0.f, 0.f, 0.f, 0.f};
#pragma unroll
    for (int ks = 0; ks < 32; ++ks) {
      const int k0 = ks * 4 + koff;
      v2f a;
      a.x = s_pexp[nloc][k0];
      a.y = s_pexp[nloc][k0 + 1];
      acc = __builtin_amdgcn_wmma_f32_16x16x4_f32(
          /*neg_a=*/false, a, /*neg_b=*/false, bfrag[ks],
          /*c_mod=*/(short)0, acc, /*reuse_a=*/false, /*reuse_b=*/false);
    }

    // phase 4: score' = (log(sum)+rowmax+h_t)*m + score*(1-m)
#pragma unroll
    for (int v = 0; v < 8; ++v) {
      const int r = v + mrow;       // tile row 0..15
      const int b = bbase + r;      // global batch row
      const float hval = h[(b * T_LEN + t) * K_TAGS + jcol];
      const float newv = __logf(fmaxf(acc[v], 1e-37f)) + s_rowmax[r] + hval;
      const float mv   = mask[b * T_LEN + t];
      const float sc   = newv * mv + oldsc[v] * (1.0f - mv);
      oldsc[v] = sc;
      s_score[r][jcol] = sc;
    }
    __syncthreads();
  }

  // ---- final: out[b] = logsumexp_j(score[b,j] + trans[EOS, j]) ----
  const float* trE = trans + 2 * K_TAGS;  // EOS_IDX = 2
#pragma unroll
  for (int rr = 0; rr < 2; ++rr) {
    const int r = w * 2 + rr;
    const float v0 = s_score[r][lane +  0] + trE[lane +  0];
    const float v1 = s_score[r][lane + 32] + trE[lane + 32];
    const float v2 = s_score[r][lane + 64] + trE[lane + 64];
    const float v3 = s_score[r][lane + 96] + trE[lane + 96];
    float mx = fmaxf(fmaxf(v0, v1), fmaxf(v2, v3));
    for (int off = 16; off > 0; off >>= 1)
      mx = fmaxf(mx, __shfl_xor(mx, off, 32));
    float s = __expf(v0 - mx) + __expf(v1 - mx) + __expf(v2 - mx) + __expf(v3 - mx);
    for (int off = 16; off > 0; off >>= 1)
      s += __shfl_xor(s, off, 32);
    if (lane == 0) out[bbase + r] = __logf(s) + mx;
  }
}

extern "C" void kernel_launch(void* const* d_in, const int* in_sizes, int n_in,
                              void* d_out, int out_size, void* d_ws, size_t ws_size,
                              hipStream_t stream) {
  (void)in_sizes; (void)n_in; (void)out_size; (void)d_ws; (void)ws_size;
  const float* h     = (const float*)d_in[0];
  const float* mask  = (const float*)d_in[1];
  const float* trans = (const float*)d_in[2];
  float* out = (float*)d_out;
  crf_forward_kernel<<<B_SZ / 16, 256, 0, stream>>>(h, mask, trans, out);
}